// LigerLinearCrossEntropyLoss_70300024701269
// MI455X (gfx1250) — compile-verified
//
#include <hip/hip_runtime.h>
#include <hip/hip_bf16.h>
#include <math.h>

// ---------------------------------------------------------------------------
// Fused linear (lm-head) + cross-entropy loss for gfx1250 (MI455X).
//   loss = mean over valid tokens of [ logsumexp_v(x_n . W_v) - x_n . W_t(n) ]
// Streams the vocab with an online logsumexp; logits are never materialized.
// Matmul runs on the CDNA5 matrix pipe via V_WMMA_F32_16X16X4_F32.
// ---------------------------------------------------------------------------

typedef __attribute__((ext_vector_type(2))) float v2f;
typedef __attribute__((ext_vector_type(8))) float v8f;

#define TOKENS   16            // WMMA M tile = tokens per block
#define D_DIM    2048
#define V_DIM    32000
#define XSTRIDE  2052          // padded LDS row stride (floats); 2052*4 = 8208 B,
                               // 16B-aligned rows, 4-bank skew -> conflict-free b64
#define WAVES    8
#define VTILE    (WAVES * 16)  // 128 vocab columns per block iteration
#define IGNORE_INDEX (-100)

__global__ void flce_init_ws(float* acc) {
    acc[0] = 0.0f;   // sum of nll
    acc[1] = 0.0f;   // valid-token count
}

__global__ __launch_bounds__(256)
void flce_main(const float* __restrict__ x,     // [N, D]
               const float* __restrict__ w,     // [V, D]
               const int*   __restrict__ tgt,   // [N]
               float* __restrict__ acc)         // ws: [loss_sum, count]
{
    __shared__ float xs[TOKENS * XSTRIDE];      // 131,328 B x-tile
    __shared__ float red_m[WAVES][TOKENS];
    __shared__ float red_s[WAVES][TOKENS];
    __shared__ float tlogit[TOKENS];
    __shared__ int   tgt_s[TOKENS];

    const int tid    = threadIdx.x;
    const int lane   = tid & 31;
    const int wave   = tid >> 5;
    const int lane15 = lane & 15;
    const int hi     = (lane >> 4) & 1;         // half-wave: K-offset 0 or 2
    const int tokenBase = blockIdx.x * TOKENS;

    // ---- stage x tile: 16 rows x 2048 f32, float4 copies, padded rows ----
    {
        const float4* xg = (const float4*)(x + (size_t)tokenBase * D_DIM);
        for (int i = tid; i < TOKENS * (D_DIM / 4); i += 256) {
            const int row = i >> 9;             // / (D_DIM/4)
            const int c4  = i & 511;
            float4 v = xg[row * (D_DIM / 4) + c4];
            *(float4*)&xs[row * XSTRIDE + c4 * 4] = v;
        }
        if (tid < TOKENS) {
            tgt_s[tid]  = tgt[tokenBase + tid];
            tlogit[tid] = 0.0f;
        }
    }
    __syncthreads();

    // A-fragment base: lane l<16 reads row l, K+{0,1}; l>=16 reads row l-16, K+{2,3}
    const float* xlane = &xs[lane15 * XSTRIDE + 2 * hi];

    float rm[8], rs[8];                          // per-lane online (max, sumexp)
    #pragma unroll
    for (int i = 0; i < 8; ++i) { rm[i] = -3.0e38f; rs[i] = 0.0f; }

    for (int vt = 0; vt < V_DIM / VTILE; ++vt) {
        const int   vcol  = vt * VTILE + wave * 16 + lane15;   // this lane's vocab id
        const float* wlane = w + (size_t)vcol * D_DIM + 2 * hi; // B frag base (same half split)

        v8f c = {};
        #pragma unroll 8
        for (int k = 0; k < D_DIM; k += 4) {
            v2f a = *(const v2f*)(xlane + k);   // ds_load_b64 (LDS broadcast per half)
            v2f b = *(const v2f*)(wlane + k);   // global_load_b64, streams W row
            c = __builtin_amdgcn_wmma_f32_16x16x4_f32(
                    false, a, false, b, (short)0, c, false, false);
        }

        // online logsumexp update; row i of C = token (i + 8*hi), column = vcol
        #pragma unroll
        for (int i = 0; i < 8; ++i) {
            const float z  = c[i];
            const float nm = fmaxf(rm[i], z);
            rs[i] = rs[i] * __expf(rm[i] - nm) + __expf(z - nm);
            rm[i] = nm;
            const int tok = i + 8 * hi;
            if (tgt_s[tok] == vcol) tlogit[tok] = z;   // exactly one lane matches
        }
    }

    // ---- merge (m,s) across the 16 lanes of each half-wave (wave32 shuffles) ----
    #pragma unroll
    for (int off = 8; off >= 1; off >>= 1) {
        #pragma unroll
        for (int i = 0; i < 8; ++i) {
            const float om = __shfl_xor(rm[i], off, 32);
            const float os = __shfl_xor(rs[i], off, 32);
            const float nm = fmaxf(rm[i], om);
            rs[i] = rs[i] * __expf(rm[i] - nm) + os * __expf(om - nm);
            rm[i] = nm;
        }
    }
    if (lane15 == 0) {                           // lanes 0 and 16 publish
        #pragma unroll
        for (int i = 0; i < 8; ++i) {
            red_m[wave][i + 8 * hi] = rm[i];
            red_s[wave][i + 8 * hi] = rs[i];
        }
    }
    __syncthreads();

    // ---- final per-token merge across waves + loss accumulation ----
    if (tid < TOKENS) {
        float m = -3.0e38f, s = 0.0f;
        #pragma unroll
        for (int wv = 0; wv < WAVES; ++wv) {
            const float om = red_m[wv][tid];
            const float os = red_s[wv][tid];
            const float nm = fmaxf(m, om);
            s = s * __expf(m - nm) + os * __expf(om - nm);
            m = nm;
        }
        const float lse  = m + __logf(s);
        const int   tg   = tgt_s[tid];
        float nll = (tg != IGNORE_INDEX) ? (lse - tlogit[tid]) : 0.0f;
        float cnt = (tg != IGNORE_INDEX) ? 1.0f : 0.0f;
        #pragma unroll
        for (int off = 8; off >= 1; off >>= 1) {
            nll += __shfl_xor(nll, off, 32);
            cnt += __shfl_xor(cnt, off, 32);
        }
        if (tid == 0) {
            atomicAdd(&acc[0], nll);
            atomicAdd(&acc[1], cnt);
        }
    }
}

__global__ void flce_finalize(const float* acc, float* out) {
    out[0] = acc[0] / fmaxf(acc[1], 1.0f);
}

extern "C" void kernel_launch(void* const* d_in, const int* in_sizes, int n_in,
                              void* d_out, int out_size, void* d_ws, size_t ws_size,
                              hipStream_t stream) {
    const float* x = (const float*)d_in[0];   // [B,S,D] f32
    const float* w = (const float*)d_in[1];   // [V,D]   f32
    const int*   t = (const int*)d_in[2];     // [B,S]   int
    float* out = (float*)d_out;
    float* acc = (float*)d_ws;

    const int N = in_sizes[2];                // B*S = 8192 tokens
    const int nblocks = N / TOKENS;           // 512

    flce_init_ws<<<1, 1, 0, stream>>>(acc);
    flce_main<<<nblocks, 256, 0, stream>>>(x, w, t, acc);
    flce_finalize<<<1, 1, 0, stream>>>(acc, out);
}